// Linear_30597347016799
// MI455X (gfx1250) — compile-verified
//
#include <hip/hip_runtime.h>
#include <hip/hip_bf16.h>

// Block-diagonal grouped linear (e3nn-style) for lmax=3, C=128, N=50000.
//   out[n, l^2:l^2+2l+1, :] = pw[l] * x[n, l^2:l^2+2l+1, :] @ W[l]
//
// Strategy: for each l, the rows {(n, m) : m in l-block} across all nodes form
// a virtual [(2l+1)*N, 128] matrix multiplied by one 128x128 weight matrix.
// All row counts are divisible by 16 -> perfect 16-row WMMA tiles.
// f32 path via V_WMMA_F32_16X16X4_F32 (full precision; kernel is ~memory bound:
// 820 MB moved vs 26.2 GFLOP -> ~32 FLOP/B).

typedef __attribute__((ext_vector_type(2))) float v2f;
typedef __attribute__((ext_vector_type(8))) float v8f;

#define NNODES 50000
#define CH 128
#define NM 16

// LDS weight layout: [k/2][n][k&1]  (64 x 128 x 2 floats = 64 KB)
// B fragment for WMMA f32 16x16x4: lane holds W[kb+2*half + {0,1}][n],
// half = lane>>4, n = lane&15  -> one 8-byte LDS read per fragment.

template <int L>
__global__ __launch_bounds__(256) void sh_linear_kernel(
    const float* __restrict__ x, const float* __restrict__ w,
    const float* __restrict__ pw, float* __restrict__ out) {
  constexpr int MR = 2 * L + 1;            // m-rows per node for this l
  constexpr int RSTART = L * L;            // first m index of this l block
  constexpr int TILES = (MR * NNODES) / 16;  // exact (all divisible by 16)

  __shared__ float wlds[64 * 128 * 2];  // 64 KB

  // Load pw[L] * W[L] into LDS, K-pair interleaved.
  const float* wl = w + (size_t)L * CH * CH;
  const float pwl = pw[L];
  for (int i = threadIdx.x; i < CH * CH; i += 256) {
    int k = i >> 7;     // weight row (c_in)
    int n = i & 127;    // weight col (c_out)
    wlds[(k >> 1) * 256 + n * 2 + (k & 1)] = wl[i] * pwl;
  }
  __syncthreads();

  const int wave = threadIdx.x >> 5;
  const int lane = threadIdx.x & 31;
  const int tile = blockIdx.x * 8 + wave;  // 16-row tile index within this l
  if (tile >= TILES) return;               // uniform per wave (EXEC stays all-1)

  const int half = lane >> 4;   // 0: K=0,1  1: K=2,3
  const int nlo = lane & 15;    // M for A, N for B/D

  // A-row for this lane: row j of the virtual [(2l+1)*N, 128] matrix.
  const int j = tile * 16 + nlo;
  const int node = j / MR;                       // const division (magic mul)
  const int m = RSTART + (j - node * MR);
  const float* xrow = x + ((size_t)node * NM + m) * CH;

  const v2f* __restrict__ wv = (const v2f*)wlds;  // v2f units

  v8f acc[8] = {};  // 8 column tiles of 16 -> full 128 out channels

#pragma unroll 2
  for (int kb = 0; kb < CH; kb += 4) {
    // A fragment: x[row][kb + 2*half + {0,1}]
    v2f a = *(const v2f*)(xrow + kb + 2 * half);
    const int kp = (kb >> 1) + half;  // K-pair row in LDS
#pragma unroll
    for (int jt = 0; jt < 8; ++jt) {
      v2f b = wv[kp * 128 + jt * 16 + nlo];
      acc[jt] = __builtin_amdgcn_wmma_f32_16x16x4_f32(
          /*neg_a=*/false, a, /*neg_b=*/false, b,
          /*c_mod=*/(short)0, acc[jt], /*reuse_a=*/false, /*reuse_b=*/false);
    }
  }

  // Store: D layout -> lane covers N = jt*16 + nlo; VGPR r -> M = r + 8*half.
#pragma unroll
  for (int r = 0; r < 8; ++r) {
    const int M = r + 8 * half;
    const int jm = tile * 16 + M;
    const int nodem = jm / MR;
    const int mm = RSTART + (jm - nodem * MR);
    float* orow = out + ((size_t)nodem * NM + mm) * CH;
#pragma unroll
    for (int jt = 0; jt < 8; ++jt) {
      orow[jt * 16 + nlo] = acc[jt][r];
    }
  }
}

extern "C" void kernel_launch(void* const* d_in, const int* in_sizes, int n_in,
                              void* d_out, int out_size, void* d_ws, size_t ws_size,
                              hipStream_t stream) {
  const float* x = (const float*)d_in[0];   // [50000, 16, 128]
  const float* w = (const float*)d_in[1];   // [4, 128, 128]
  const float* pw = (const float*)d_in[2];  // [4]
  float* out = (float*)d_out;               // [50000, 16, 128]

  // blocks = ceil(tiles_l / 8), tiles_l = (2l+1)*50000/16
  sh_linear_kernel<0><<<391, 256, 0, stream>>>(x, w, pw, out);   // 3125 tiles
  sh_linear_kernel<1><<<1172, 256, 0, stream>>>(x, w, pw, out);  // 9375 tiles
  sh_linear_kernel<2><<<1954, 256, 0, stream>>>(x, w, pw, out);  // 15625 tiles
  sh_linear_kernel<3><<<2735, 256, 0, stream>>>(x, w, pw, out);  // 21875 tiles
}